// MultiplicativeMLPLayer_67619965108914
// MI455X (gfx1250) — compile-verified
//
#include <hip/hip_runtime.h>
#include <math.h>

// ---------------- CDNA5 WMMA types ----------------
typedef float v2f __attribute__((ext_vector_type(2)));
typedef float v8f __attribute__((ext_vector_type(8)));

#define D_      512
#define TOTAL_  4096
#define TOPK_   8
#define NROWS_  2048      // T * TOPK = 256*8
#define MAXH_   2560
#define NTOK_   256

#define WMMA_(a, b, c)                                                        \
  __builtin_amdgcn_wmma_f32_16x16x4_f32(false, a, false, b, (short)0, c,      \
                                        false, false)

// activation: 0 gelu(exact), 1 relu, 2 tanh, 3 silu
__device__ __forceinline__ float act_apply(int a, float v) {
  switch (a & 3) {
    case 0:  return 0.5f * v * (1.0f + erff(v * 0.7071067811865475f));
    case 1:  return fmaxf(v, 0.0f);
    case 2:  return tanhf(v);
    default: return v / (1.0f + expf(-v));   // silu = x*sigmoid(x)
  }
}

// 8 independent WMMAs of one 32x64 k-step (2 M-subtiles x 4 N-subtiles)
#define COMPUTE8(A0v, A1v, B0v, B1v, B2v, B3v)                                \
  {                                                                           \
    c00 = WMMA_(A0v, B0v, c00);                                               \
    c01 = WMMA_(A0v, B1v, c01);                                               \
    c02 = WMMA_(A0v, B2v, c02);                                               \
    c03 = WMMA_(A0v, B3v, c03);                                               \
    c10 = WMMA_(A1v, B0v, c10);                                               \
    c11 = WMMA_(A1v, B1v, c11);                                               \
    c12 = WMMA_(A1v, B2v, c12);                                               \
    c13 = WMMA_(A1v, B3v, c13);                                               \
  }

// =====================================================================
// Dense GEMM, 32x64 tile per wave, 2-stage software pipeline:
//   C = act(A @ B + bias)
// A: MxK row-major, B: KxN row-major. K mult of 32, N mult of 64, M of 32.
// WMMA f32 16x16x4 fragment layout (ISA 7.12.2):
//   A 16x4 : lanes 0-15 -> M=lane, VGPR0=K0 VGPR1=K1 ; lanes 16-31 -> K2,K3
//   B 4x16 : VGPR0 lanes0-15 K=0 / lanes16-31 K=2 ; VGPR1 K=1 / K=3
//   C/D    : VGPR j -> row j (lanes 0-15) and row j+8 (lanes 16-31)
// =====================================================================
__global__ void __launch_bounds__(32, 1)
dense_gemm_bias_act(const float* __restrict__ A,
                    const float* __restrict__ B,
                    const float* __restrict__ bias,
                    float* __restrict__ C,
                    int N, int K, int act) {
  const int lane = threadIdx.x;            // blockDim.x == 32 (wave32)
  const int n0   = blockIdx.x * 64;
  const int m0   = blockIdx.y * 32;
  const int kh2  = (lane >> 4) * 2;        // 0 or 2
  const int nc   = lane & 15;
  const float* aPtr0 = A + (long)(m0 + (lane & 15)) * K + kh2;
  const float* aPtr1 = aPtr0 + (long)16 * K;
  const float* bPtr  = B + (long)kh2 * N + n0 + nc;
  const long  row4 = (long)4 * N;          // advance 4 K-rows

  // warm L2/L1 ahead of the streaming loop (speculative, safe)
  __builtin_prefetch(bPtr + (long)8 * row4, 0, 1);
  __builtin_prefetch(bPtr + (long)16 * row4, 0, 1);

  v8f c00 = {}, c01 = {}, c02 = {}, c03 = {};
  v8f c10 = {}, c11 = {}, c12 = {}, c13 = {};

#define DLOAD(A0v, A1v, B0v, B1v, B2v, B3v)                                   \
  {                                                                           \
    const float* br_ = bPtr + N;                                              \
    A0v.x = aPtr0[0]; A0v.y = aPtr0[1];                                       \
    A1v.x = aPtr1[0]; A1v.y = aPtr1[1];                                       \
    B0v.x = bPtr[0];  B0v.y = br_[0];                                         \
    B1v.x = bPtr[16]; B1v.y = br_[16];                                        \
    B2v.x = bPtr[32]; B2v.y = br_[32];                                        \
    B3v.x = bPtr[48]; B3v.y = br_[48];                                        \
    aPtr0 += 4; aPtr1 += 4; bPtr += row4;                                     \
  }

  v2f A0, A1, B0, B1, B2, B3;
  v2f nA0, nA1, nB0, nB1, nB2, nB3;
  DLOAD(A0, A1, B0, B1, B2, B3)
#pragma unroll 2
  for (int k = 4; k < K; k += 4) {
    DLOAD(nA0, nA1, nB0, nB1, nB2, nB3)                // next step's loads
    COMPUTE8(A0, A1, B0, B1, B2, B3)                   // current step's math
    A0 = nA0; A1 = nA1; B0 = nB0; B1 = nB1; B2 = nB2; B3 = nB3;
  }
  COMPUTE8(A0, A1, B0, B1, B2, B3)
#undef DLOAD

#define DENSE_STORE(cc, mh, sub)                                              \
  {                                                                           \
    const int mrow = m0 + (mh)*16 + (lane >> 4) * 8;                          \
    const int col = n0 + (sub)*16 + nc;                                       \
    const float bv = bias[col];                                               \
    _Pragma("unroll") for (int j = 0; j < 8; ++j) {                           \
      float v = cc[j] + bv;                                                   \
      if (act >= 0) v = act_apply(act, v);                                    \
      C[(long)(mrow + j) * N + col] = v;                                      \
    }                                                                         \
  }
  DENSE_STORE(c00, 0, 0) DENSE_STORE(c01, 0, 1)
  DENSE_STORE(c02, 0, 2) DENSE_STORE(c03, 0, 3)
  DENSE_STORE(c10, 1, 0) DENSE_STORE(c11, 1, 1)
  DENSE_STORE(c12, 1, 2) DENSE_STORE(c13, 1, 3)
#undef DENSE_STORE
}

// =====================================================================
// Grouped (MoE) GEMM over compacted row lists, 32x64 tile per wave,
// 2-stage software pipeline.
//   e = blockIdx.z, rows = lists[(kind*16+e)*2048 .. +counts]
//   A-row for compacted row r is (r >> arowShift)   (3 for pre: token id)
//   K = useHeK ? he(e) : Kfixed ;  N = useHeN ? he(e) : Nfixed
//   actMode: -1 none, -2 per-expert (e&3), >=0 fixed
// =====================================================================
__global__ void __launch_bounds__(32, 1)
grouped_gemm(const float* __restrict__ A, long lda, int arowShift,
             const float* __restrict__ Bb, long bStrideE, long ldb,
             const float* __restrict__ biasB, long biasStrideE,
             float* __restrict__ Cb, long ldc,
             const int* __restrict__ lists,
             const int* __restrict__ counts, int kind,
             int Kfixed, int useHeK, int Nfixed, int useHeN,
             int actMode) {
  const int e   = blockIdx.z;
  const int cnt = counts[kind * 16 + e];
  const int mt  = blockIdx.y;
  if (mt * 32 >= cnt) return;                       // block-uniform exit
  const int he = 512 * (2 + (e >> 2));
  const int N  = useHeN ? he : Nfixed;
  const int n0 = blockIdx.x * 64;
  if (n0 >= N) return;                              // block-uniform exit
  const int K  = useHeK ? he : Kfixed;
  const int* list = lists + (long)(kind * 16 + e) * NROWS_;

  const int lane = threadIdx.x;
  const int kh2  = (lane >> 4) * 2;
  const int nc   = lane & 15;
  const int li0  = mt * 32 + (lane & 15);
  const int li1  = li0 + 16;
  const int rA0  = (li0 < cnt) ? (list[li0] >> arowShift) : 0;
  const int rA1  = (li1 < cnt) ? (list[li1] >> arowShift) : 0;
  const float msk0 = (li0 < cnt) ? 1.0f : 0.0f;
  const float msk1 = (li1 < cnt) ? 1.0f : 0.0f;

  const float* aPtr0 = A + (long)rA0 * lda + kh2;
  const float* aPtr1 = A + (long)rA1 * lda + kh2;
  const float* bPtr  = Bb + (long)e * bStrideE + (long)kh2 * ldb + n0 + nc;
  const long  row4 = (long)4 * ldb;

  __builtin_prefetch(bPtr + (long)8 * row4, 0, 1);
  __builtin_prefetch(bPtr + (long)16 * row4, 0, 1);

  v8f c00 = {}, c01 = {}, c02 = {}, c03 = {};
  v8f c10 = {}, c11 = {}, c12 = {}, c13 = {};

#define GLOAD(A0v, A1v, B0v, B1v, B2v, B3v)                                   \
  {                                                                           \
    const float* br_ = bPtr + ldb;                                            \
    A0v.x = aPtr0[0] * msk0; A0v.y = aPtr0[1] * msk0;                         \
    A1v.x = aPtr1[0] * msk1; A1v.y = aPtr1[1] * msk1;                         \
    B0v.x = bPtr[0];  B0v.y = br_[0];                                         \
    B1v.x = bPtr[16]; B1v.y = br_[16];                                        \
    B2v.x = bPtr[32]; B2v.y = br_[32];                                        \
    B3v.x = bPtr[48]; B3v.y = br_[48];                                        \
    aPtr0 += 4; aPtr1 += 4; bPtr += row4;                                     \
  }

  v2f A0, A1, B0, B1, B2, B3;
  v2f nA0, nA1, nB0, nB1, nB2, nB3;
  GLOAD(A0, A1, B0, B1, B2, B3)
#pragma unroll 2
  for (int k = 4; k < K; k += 4) {
    GLOAD(nA0, nA1, nB0, nB1, nB2, nB3)
    COMPUTE8(A0, A1, B0, B1, B2, B3)
    A0 = nA0; A1 = nA1; B0 = nB0; B1 = nB1; B2 = nB2; B3 = nB3;
  }
  COMPUTE8(A0, A1, B0, B1, B2, B3)
#undef GLOAD

  const float* bias = biasB + (long)e * biasStrideE;
  const int act = (actMode == -2) ? (e & 3) : actMode;
#define GROUP_STORE(cc, mh, sub)                                              \
  {                                                                           \
    const int col = n0 + (sub)*16 + nc;                                       \
    const float bv = bias[col];                                               \
    _Pragma("unroll") for (int j = 0; j < 8; ++j) {                           \
      const int lj = mt * 32 + (mh)*16 + j + (lane >> 4) * 8;                 \
      if (lj < cnt) {                                                         \
        const int r = list[lj];                                               \
        float v = cc[j] + bv;                                                 \
        if (act >= 0) v = act_apply(act, v);                                  \
        Cb[(long)r * ldc + col] = v;                                          \
      }                                                                       \
    }                                                                         \
  }
  GROUP_STORE(c00, 0, 0) GROUP_STORE(c01, 0, 1)
  GROUP_STORE(c02, 0, 2) GROUP_STORE(c03, 0, 3)
  GROUP_STORE(c10, 1, 0) GROUP_STORE(c11, 1, 1)
  GROUP_STORE(c12, 1, 2) GROUP_STORE(c13, 1, 3)
#undef GROUP_STORE
}

// ---------------- block reduction helper (256 threads) ----------------
__device__ __forceinline__ float block_sum256(float v, float* sred, int tid) {
  sred[tid] = v;
  __syncthreads();
  for (int off = 128; off > 0; off >>= 1) {
    if (tid < off) sred[tid] += sred[tid + off];
    __syncthreads();
  }
  float r = sred[0];
  __syncthreads();
  return r;
}

// =====================================================================
// Softmax (over 4096, with temperature) + iterative top-8 argmax.
// Tie-break: lowest index first (matches jax.lax.top_k).
// One 256-thread block per token.
// =====================================================================
__global__ void softmax_topk(const float* __restrict__ scores,
                             const float* __restrict__ temp,
                             float* __restrict__ topw,
                             int* __restrict__ preidx,
                             int* __restrict__ mlpidx,
                             int* __restrict__ postidx) {
  __shared__ float sv[256];
  __shared__ int   si[256];
  __shared__ int   chosen[TOPK_];
  __shared__ float s_mx, s_den;
  const int t = blockIdx.x, tid = threadIdx.x;
  const float* s = scores + (long)t * TOTAL_;
  const float invT = 1.0f / temp[0];

  // max of s*invT
  float m = -INFINITY;
  for (int i = tid; i < TOTAL_; i += 256) m = fmaxf(m, s[i] * invT);
  sv[tid] = m;
  __syncthreads();
  for (int off = 128; off > 0; off >>= 1) {
    if (tid < off) sv[tid] = fmaxf(sv[tid], sv[tid + off]);
    __syncthreads();
  }
  if (tid == 0) s_mx = sv[0];
  __syncthreads();
  const float mx = s_mx;
  __syncthreads();

  // softmax denominator
  float sum = 0.0f;
  for (int i = tid; i < TOTAL_; i += 256) sum += expf(s[i] * invT - mx);
  sum = block_sum256(sum, sv, tid);
  if (tid == 0) s_den = sum;
  __syncthreads();
  const float den = s_den;
  __syncthreads();

  // iterative top-8
  for (int kk = 0; kk < TOPK_; ++kk) {
    float bv = -INFINITY;
    int   bi = 1 << 30;
    for (int i = tid; i < TOTAL_; i += 256) {
      bool taken = false;
      for (int j = 0; j < kk; ++j) taken |= (chosen[j] == i);
      if (taken) continue;
      const float v = s[i] * invT;
      if (v > bv || (v == bv && i < bi)) { bv = v; bi = i; }
    }
    sv[tid] = bv; si[tid] = bi;
    __syncthreads();
    for (int off = 128; off > 0; off >>= 1) {
      if (tid < off) {
        if (sv[tid + off] > sv[tid] ||
            (sv[tid + off] == sv[tid] && si[tid + off] < si[tid])) {
          sv[tid] = sv[tid + off];
          si[tid] = si[tid + off];
        }
      }
      __syncthreads();
    }
    if (tid == 0) {
      const int idx = si[0];
      chosen[kk] = idx;
      const float prob = expf(sv[0] - mx) / den;
      topw[t * TOPK_ + kk] = (prob >= 1e-6f) ? prob : 0.0f;
      const int r = t * TOPK_ + kk;
      preidx[r]  = idx / 256;          // idx // (NMLP*NPOST)
      const int rem = idx % 256;
      mlpidx[r]  = rem / 16;
      postidx[r] = rem % 16;
    }
    __syncthreads();
  }
}

// per-(kind,expert) row-list compaction; deterministic serial scan
__global__ void build_lists(const int* __restrict__ preidx,
                            const int* __restrict__ mlpidx,
                            const int* __restrict__ postidx,
                            int* __restrict__ lists, int* __restrict__ counts) {
  const int b = blockIdx.x;                       // 0..47 : kind*16+e
  if (threadIdx.x != 0) return;
  const int kind = b / 16, e = b % 16;
  const int* idx = (kind == 0) ? preidx : (kind == 1) ? mlpidx : postidx;
  int* list = lists + (long)b * NROWS_;
  int c = 0;
  for (int r = 0; r < NROWS_; ++r)
    if (idx[r] == e) list[c++] = r;
  counts[b] = c;
}

// pre stage: per-row layernorm + per-expert activation, in place on xin
__global__ void pre_ln_act(float* __restrict__ xin,
                           const int* __restrict__ preidx,
                           const float* __restrict__ g,
                           const float* __restrict__ be) {
  __shared__ float sred[256];
  const int r = blockIdx.x, tid = threadIdx.x;
  const int e = preidx[r];
  float* row = xin + (long)r * D_;
  const float v0 = row[tid], v1 = row[tid + 256];
  const float mean = block_sum256(v0 + v1, sred, tid) * (1.0f / D_);
  const float d0 = v0 - mean, d1 = v1 - mean;
  const float var = block_sum256(d0 * d0 + d1 * d1, sred, tid) * (1.0f / D_);
  const float inv = rsqrtf(var + 1e-5f);
  const int a = e & 3;
  const long gb = (long)e * D_;
  row[tid]       = act_apply(a, d0 * inv * g[gb + tid]       + be[gb + tid]);
  row[tid + 256] = act_apply(a, d1 * inv * g[gb + tid + 256] + be[gb + tid + 256]);
}

// post stage: conditional layernorm (even experts) + weighted top-k reduce
__global__ void post_finalize(const float* __restrict__ postraw,
                              const int* __restrict__ postidx,
                              const float* __restrict__ topw,
                              const float* __restrict__ g,
                              const float* __restrict__ be,
                              float* __restrict__ out) {
  __shared__ float sred[256];
  const int t = blockIdx.x, tid = threadIdx.x;
  float acc0 = 0.0f, acc1 = 0.0f;
  for (int k = 0; k < TOPK_; ++k) {
    const int r = t * TOPK_ + k;
    const int e = postidx[r];              // uniform across the block
    const float* z = postraw + (long)r * D_;
    float v0 = z[tid], v1 = z[tid + 256];
    if ((e & 1) == 0) {
      const float mean = block_sum256(v0 + v1, sred, tid) * (1.0f / D_);
      const float d0 = v0 - mean, d1 = v1 - mean;
      const float var = block_sum256(d0 * d0 + d1 * d1, sred, tid) * (1.0f / D_);
      const float inv = rsqrtf(var + 1e-5f);
      const long gb = (long)e * D_;
      v0 = d0 * inv * g[gb + tid]       + be[gb + tid];
      v1 = d1 * inv * g[gb + tid + 256] + be[gb + tid + 256];
    }
    const float w = topw[t * TOPK_ + k];
    acc0 += w * v0;
    acc1 += w * v1;
  }
  out[(long)t * D_ + tid]       = acc0;
  out[(long)t * D_ + tid + 256] = acc1;
}

// =====================================================================
extern "C" void kernel_launch(void* const* d_in, const int* in_sizes, int n_in,
                              void* d_out, int out_size, void* d_ws, size_t ws_size,
                              hipStream_t stream) {
  const float* x     = (const float*)d_in[0];
  const float* r_w1  = (const float*)d_in[1];
  const float* r_b1  = (const float*)d_in[2];
  const float* r_w2  = (const float*)d_in[3];
  const float* r_b2  = (const float*)d_in[4];
  const float* r_w3  = (const float*)d_in[5];
  const float* r_b3  = (const float*)d_in[6];
  const float* temp  = (const float*)d_in[7];
  const float* pre_w = (const float*)d_in[8];
  const float* pre_b = (const float*)d_in[9];
  const float* pre_g = (const float*)d_in[10];
  const float* pre_be= (const float*)d_in[11];
  const float* mlp_w1= (const float*)d_in[12];
  const float* mlp_b1= (const float*)d_in[13];
  const float* mlp_w2= (const float*)d_in[14];
  const float* mlp_b2= (const float*)d_in[15];
  const float* post_w= (const float*)d_in[16];
  const float* post_b= (const float*)d_in[17];
  const float* post_g= (const float*)d_in[18];
  const float* post_be=(const float*)d_in[19];
  float* out = (float*)d_out;

  // ---- workspace layout (bytes) ----
  char* ws = (char*)d_ws;
  size_t off = 0;
  auto alloc = [&](size_t bytes) { char* p = ws + off; off = (off + bytes + 255) & ~(size_t)255; return p; };
  float* h1      = (float*)alloc((size_t)NTOK_ * 512 * 4);    // 256x512
  float* h2      = (float*)alloc((size_t)NTOK_ * 256 * 4);    // 256x256
  float* scores  = (float*)alloc((size_t)NTOK_ * TOTAL_ * 4); // 256x4096
  float* topw    = (float*)alloc((size_t)NTOK_ * TOPK_ * 4);
  int*   preidx  = (int*)  alloc((size_t)NROWS_ * 4);
  int*   mlpidx  = (int*)  alloc((size_t)NROWS_ * 4);
  int*   postidx = (int*)  alloc((size_t)NROWS_ * 4);
  int*   counts  = (int*)  alloc(48 * 4);
  int*   lists   = (int*)  alloc((size_t)48 * NROWS_ * 4);
  float* xin     = (float*)alloc((size_t)NROWS_ * D_ * 4);    // 2048x512
  float* hmlp    = (float*)alloc((size_t)NROWS_ * MAXH_ * 4); // 2048x2560
  float* mlpout  = (float*)alloc((size_t)NROWS_ * D_ * 4);
  float* postraw = (float*)alloc((size_t)NROWS_ * D_ * 4);
  (void)in_sizes; (void)n_in; (void)out_size; (void)ws_size;

  const dim3 wave(32);

  // ---- router ----
  dense_gemm_bias_act<<<dim3(512 / 64, NTOK_ / 32), wave, 0, stream>>>(
      x, r_w1, r_b1, h1, 512, 512, /*gelu*/0);
  dense_gemm_bias_act<<<dim3(256 / 64, NTOK_ / 32), wave, 0, stream>>>(
      h1, r_w2, r_b2, h2, 256, 512, /*gelu*/0);
  dense_gemm_bias_act<<<dim3(TOTAL_ / 64, NTOK_ / 32), wave, 0, stream>>>(
      h2, r_w3, r_b3, scores, TOTAL_, 256, /*none*/-1);

  softmax_topk<<<NTOK_, 256, 0, stream>>>(scores, temp, topw, preidx, mlpidx, postidx);
  build_lists<<<48, 32, 0, stream>>>(preidx, mlpidx, postidx, lists, counts);

  // ---- pre experts: xin = x[token] @ pre_w[e] + pre_b[e] (LN+act after) ----
  grouped_gemm<<<dim3(512 / 64, NROWS_ / 32, 16), wave, 0, stream>>>(
      x, 512, /*arowShift: row->token*/3,
      pre_w, (long)512 * 512, 512,
      pre_b, 512,
      xin, 512,
      lists, counts, /*kind*/0,
      /*K*/512, 0, /*N*/512, 0, /*act*/-1);
  pre_ln_act<<<NROWS_, 256, 0, stream>>>(xin, preidx, pre_g, pre_be);

  // ---- MLP layer 1: hmlp = act_e(xin @ mlp_w1[e][:, :he] + b1) ----
  grouped_gemm<<<dim3(MAXH_ / 64, NROWS_ / 32, 16), wave, 0, stream>>>(
      xin, 512, 0,
      mlp_w1, (long)512 * MAXH_, MAXH_,
      mlp_b1, MAXH_,
      hmlp, MAXH_,
      lists, counts, /*kind*/1,
      /*K*/512, 0, /*N=he*/0, 1, /*act: e%4*/-2);

  // ---- MLP layer 2: mlpout = hmlp[:, :he] @ mlp_w2[e][:he, :] + b2 ----
  grouped_gemm<<<dim3(512 / 64, NROWS_ / 32, 16), wave, 0, stream>>>(
      hmlp, MAXH_, 0,
      mlp_w2, (long)MAXH_ * 512, 512,
      mlp_b2, 512,
      mlpout, 512,
      lists, counts, /*kind*/1,
      /*K=he*/0, 1, /*N*/512, 0, /*act*/-1);

  // ---- post experts: postraw = mlpout @ post_w[e] + post_b[e] ----
  grouped_gemm<<<dim3(512 / 64, NROWS_ / 32, 16), wave, 0, stream>>>(
      mlpout, 512, 0,
      post_w, (long)512 * 512, 512,
      post_b, 512,
      postraw, 512,
      lists, counts, /*kind*/2,
      /*K*/512, 0, /*N*/512, 0, /*act*/-1);

  // ---- conditional LN + weighted top-k reduction ----
  post_finalize<<<NTOK_, 256, 0, stream>>>(postraw, postidx, topw,
                                           post_g, post_be, out);
}